// Point_Globalfea_31447750541863
// MI455X (gfx1250) — compile-verified
//
#include <hip/hip_runtime.h>

#define N_POINTS  1000000
#define NUM_SEG   100000
#define D_IN      64
#define D_OUT     32
#define NUM_TILES (NUM_SEG / 16)   // 6250, exact
#define CHUNK     512
#define GROUPS    4
#define TILES_PER_BLOCK 8

typedef __attribute__((ext_vector_type(2))) float v2f;
typedef __attribute__((ext_vector_type(8))) float v8f;

#define ENC_NEG_INF 0x007FFFFFu   // enc(-inf)

// Order-preserving f32 <-> u32 mapping so integer atomic max == float max.
// Branchless: 3 VALU ops each, no EXEC divergence.
__device__ __forceinline__ unsigned enc_f32(float f) {
  unsigned u = __float_as_uint(f);
  unsigned s = (unsigned)((int)u >> 31);        // all-ones if negative
  return u ^ (s | 0x80000000u);                 // neg: ~u ; pos: u|sign
}
__device__ __forceinline__ float dec_f32(unsigned e) {
  unsigned s = (unsigned)((int)e >> 31);        // all-ones if top bit set
  return __uint_as_float(e ^ (~s | 0x80000000u)); // top: e^sign ; else: ~e
}

// ---------------------------------------------------------------- init pool
__global__ __launch_bounds__(256) void init_pool(unsigned* __restrict__ pool, int n) {
  int i = blockIdx.x * 256 + threadIdx.x;
  if (i < n) pool[i] = ENC_NEG_INF;
}

// ------------------------------------------- gather + add + segment-max
// 256 threads = 4 groups of 64; group owns one 512-point chunk; thread owns
// one of the 64 dims. pt_inv is sorted, so run-length accumulate the max in a
// register and flush with one u32 atomic-max at segment / chunk boundaries.
__global__ __launch_bounds__(256) void gather_segmax(
    const int*   __restrict__ shuffled_ind,
    const float* __restrict__ gfea,
    const float* __restrict__ lfea,
    const int*   __restrict__ seg_ids,
    unsigned*    __restrict__ pool) {
  const int d   = threadIdx.x & 63;
  const int grp = threadIdx.x >> 6;
  const int p0  = (blockIdx.x * GROUPS + grp) * CHUNK;
  if (p0 >= N_POINTS) return;
  const int p1 = (p0 + CHUNK < N_POINTS) ? p0 + CHUNK : N_POINTS;

  int   seg_cur = -1;
  float run     = 0.0f;
  for (int r = p0; r < p1; ++r) {
    const int seg = seg_ids[r];        // uniform address -> scalar load
    const int gi  = shuffled_ind[r];   // uniform address -> scalar load
    const float v = gfea[(size_t)gi * D_IN + d] + lfea[(size_t)r * D_IN + d];
    if (seg != seg_cur) {              // wave-uniform branch
      if (seg_cur >= 0)
        atomicMax(&pool[(size_t)seg_cur * D_IN + d], enc_f32(run));
      seg_cur = seg;
      run = v;
    } else {
      run = fmaxf(run, v);
    }
  }
  if (seg_cur >= 0)
    atomicMax(&pool[(size_t)seg_cur * D_IN + d], enc_f32(run));
}

// ------------------------------------------------- pooled @ W + b, relu
// One wave computes one 16-row output tile via V_WMMA_F32_16X16X4_F32.
// 8 waves/block share W (64x32 f32 = 8KB) in LDS.
__global__ __launch_bounds__(256) void pooled_gemm(
    const unsigned* __restrict__ pool,
    const float*    __restrict__ W,
    const float*    __restrict__ bias,
    float*          __restrict__ out) {
  __shared__ float sW[D_IN * D_OUT];
  {
    const float4* src = (const float4*)W;
    float4*       dst = (float4*)sW;
    for (int i = threadIdx.x; i < (D_IN * D_OUT) / 4; i += 256) dst[i] = src[i];
  }
  __syncthreads();

  const int wave = threadIdx.x >> 5;
  const int tile = blockIdx.x * TILES_PER_BLOCK + wave;
  if (tile >= NUM_TILES) return;

  const int lane  = threadIdx.x & 31;
  const int hi    = lane >> 4;      // which half of the wave
  const int l15   = lane & 15;      // A row / B,C,D column within tile
  const int mBase = tile * 16;

  // Bias folds into accumulator init: C/D column == lane&15 for all 8 VGPRs.
  const float b0v = bias[l15];
  const float b1v = bias[l15 + 16];
  v8f acc0, acc1;
  for (int j = 0; j < 8; ++j) { acc0[j] = b0v; acc1[j] = b1v; }

  const unsigned* arow = pool + (size_t)(mBase + l15) * D_IN;

  for (int k = 0; k < 16; ++k) {
    const int kbase = 4 * k + 2 * hi;
    // A fragment: lane<16 holds K={4k,4k+1}, lane>=16 holds K={4k+2,4k+3}.
    uint2 ae = *(const uint2*)(arow + kbase);   // 8B aligned
    const float fx = dec_f32(ae.x);
    const float fy = dec_f32(ae.y);
    v2f a;
    a.x = (ae.x != ENC_NEG_INF) ? fx : 0.0f;    // empty segment -> 0 (cndmask)
    a.y = (ae.y != ENC_NEG_INF) ? fy : 0.0f;
    // B fragments (4x16): VGPR v holds row K=kbase+v, column = lane&15.
    v2f b0, b1;
    b0.x = sW[(kbase + 0) * D_OUT + l15];
    b0.y = sW[(kbase + 1) * D_OUT + l15];
    b1.x = sW[(kbase + 0) * D_OUT + l15 + 16];
    b1.y = sW[(kbase + 1) * D_OUT + l15 + 16];

    acc0 = __builtin_amdgcn_wmma_f32_16x16x4_f32(false, a, false, b0,
                                                 (short)0, acc0, false, false);
    acc1 = __builtin_amdgcn_wmma_f32_16x16x4_f32(false, a, false, b1,
                                                 (short)0, acc1, false, false);
  }

  // D layout: VGPR j -> row mBase + j + 8*hi, column l15 (+16 for acc1).
  for (int j = 0; j < 8; ++j) {
    const int row = mBase + j + 8 * hi;
    out[(size_t)row * D_OUT + l15]      = fmaxf(acc0[j], 0.0f);
    out[(size_t)row * D_OUT + l15 + 16] = fmaxf(acc1[j], 0.0f);
  }
}

// ----------------------------------------------------------------- launch
extern "C" void kernel_launch(void* const* d_in, const int* in_sizes, int n_in,
                              void* d_out, int out_size, void* d_ws, size_t ws_size,
                              hipStream_t stream) {
  const int*   shuffled_ind = (const int*)  d_in[0];
  const float* gfea         = (const float*)d_in[1];
  const float* lfea         = (const float*)d_in[2];
  const int*   seg_ids      = (const int*)  d_in[3];
  const float* W            = (const float*)d_in[4];
  const float* bias         = (const float*)d_in[5];
  float*       out          = (float*)d_out;
  unsigned*    pool         = (unsigned*)d_ws;   // 100000*64*4 = 25.6 MB

  const int pn = NUM_SEG * D_IN;
  init_pool<<<(pn + 255) / 256, 256, 0, stream>>>(pool, pn);

  const int nblk = (N_POINTS + CHUNK * GROUPS - 1) / (CHUNK * GROUPS);
  gather_segmax<<<nblk, 256, 0, stream>>>(shuffled_ind, gfea, lfea, seg_ids, pool);

  const int gblk = (NUM_TILES + TILES_PER_BLOCK - 1) / TILES_PER_BLOCK;
  pooled_gemm<<<gblk, 256, 0, stream>>>(pool, W, bias, out);
}